// PyG_GCN_52716428591833
// MI455X (gfx1250) — compile-verified
//
#include <hip/hip_runtime.h>

#define N_NODES 50000
#define N_EDGES 800000
#define D 64

typedef __attribute__((ext_vector_type(2))) float v2f;
typedef __attribute__((ext_vector_type(8))) float v8f;

// ---------------------------------------------------------------------------
// 1) deg[i] = 1.0 (self loop)
// ---------------------------------------------------------------------------
__global__ void gcn_init_deg(float* __restrict__ deg) {
    int i = blockIdx.x * blockDim.x + threadIdx.x;
    if (i < N_NODES) deg[i] = 1.0f;
}

// ---------------------------------------------------------------------------
// 2) deg[col[e]] += 1
// ---------------------------------------------------------------------------
__global__ void gcn_accum_deg(const int* __restrict__ col, float* __restrict__ deg) {
    int e = blockIdx.x * blockDim.x + threadIdx.x;
    if (e < N_EDGES) atomicAdd(deg + col[e], 1.0f);
}

// ---------------------------------------------------------------------------
// 3) dinv[i] = rsqrt(deg[i])
// ---------------------------------------------------------------------------
__global__ void gcn_dinv(const float* __restrict__ deg, float* __restrict__ dinv) {
    int i = blockIdx.x * blockDim.x + threadIdx.x;
    if (i < N_NODES) {
        float d = deg[i];
        dinv[i] = (d > 0.0f) ? rsqrtf(d) : 0.0f;
    }
}

// ---------------------------------------------------------------------------
// 4) x = features @ W  via V_WMMA_F32_16X16X4_F32.
//    One wave (32 lanes) computes a 16x64 tile of x.
//    A-frag (16x4 f32): lanes 0-15 hold K=4k,4k+1 ; lanes 16-31 hold K=4k+2,4k+3
//    B-frag (4x16 f32): mirrored; C/D: 8 VGPRs, lanes16-31 = rows M+8.
// ---------------------------------------------------------------------------
__global__ __launch_bounds__(256) void gcn_gemm_wmma(const float* __restrict__ F,
                                                     const float* __restrict__ W,
                                                     float* __restrict__ X) {
    const int lane    = threadIdx.x & 31;
    const int waveId  = threadIdx.x >> 5;                 // 0..7
    const int rowBlk  = blockIdx.x * 8 + waveId;          // 16 rows per wave
    if (rowBlk >= N_NODES / 16) return;                   // wave-uniform exit

    const int half = lane >> 4;                            // 0 or 1
    const int l16  = lane & 15;
    const int aRow = rowBlk * 16 + l16;

    #pragma unroll
    for (int nt = 0; nt < 4; ++nt) {
        v8f acc = {};
        const int nCol = nt * 16 + l16;
        #pragma unroll
        for (int k = 0; k < 16; ++k) {
            const int k0 = k * 4 + half * 2;
            v2f a, b;
            a.x = F[aRow * D + k0];
            a.y = F[aRow * D + k0 + 1];
            b.x = W[k0 * D + nCol];
            b.y = W[(k0 + 1) * D + nCol];
            acc = __builtin_amdgcn_wmma_f32_16x16x4_f32(
                /*neg_a=*/false, a, /*neg_b=*/false, b,
                /*c_mod=*/(short)0, acc, /*reuse_a=*/false, /*reuse_b=*/false);
        }
        const int outRowBase = rowBlk * 16 + half * 8;
        #pragma unroll
        for (int i = 0; i < 8; ++i) {
            X[(outRowBase + i) * D + nt * 16 + l16] = acc[i];
        }
    }
}

// ---------------------------------------------------------------------------
// 5) out[i][d] = b[d] + x[i][d] * dinv[i]^2   (bias + self-loop message)
// ---------------------------------------------------------------------------
__global__ void gcn_self(const float* __restrict__ X, const float* __restrict__ dinv,
                         const float* __restrict__ bias, float* __restrict__ out) {
    int t = blockIdx.x * blockDim.x + threadIdx.x;
    if (t < N_NODES * D) {
        int i = t >> 6;            // node
        int d = t & 63;            // feature
        float di = dinv[i];
        out[t] = bias[d] + X[t] * (di * di);
    }
}

// ---------------------------------------------------------------------------
// 6) out[col[e]] += x[row[e]] * dinv[row]*dinv[col]
//    16 lanes per edge, float4 gather + 4 fp32 global atomics per lane.
//    x (12.8 MB) is L2-resident on MI455X (192 MB L2) -> L2-rate scatter.
// ---------------------------------------------------------------------------
__global__ void gcn_scatter(const int* __restrict__ row, const int* __restrict__ col,
                            const float* __restrict__ X, const float* __restrict__ dinv,
                            float* __restrict__ out) {
    int t = blockIdx.x * blockDim.x + threadIdx.x;
    int e = t >> 4;
    if (e >= N_EDGES) return;
    int d = (t & 15) << 2;
    int r = row[e];
    int c = col[e];
    float norm = dinv[r] * dinv[c];
    const float4 xv = *(const float4*)(X + r * D + d);
    float* o = out + c * D + d;
    atomicAdd(o + 0, xv.x * norm);
    atomicAdd(o + 1, xv.y * norm);
    atomicAdd(o + 2, xv.z * norm);
    atomicAdd(o + 3, xv.w * norm);
}

// ---------------------------------------------------------------------------
extern "C" void kernel_launch(void* const* d_in, const int* in_sizes, int n_in,
                              void* d_out, int out_size, void* d_ws, size_t ws_size,
                              hipStream_t stream) {
    const float* features = (const float*)d_in[0];   // [50000,64]
    const float* W        = (const float*)d_in[1];   // [64,64]
    const float* bias     = (const float*)d_in[2];   // [64]
    const int*   edges    = (const int*)d_in[3];     // [2,800000] flat: row then col

    float* out  = (float*)d_out;
    float* ws   = (float*)d_ws;
    float* deg  = ws;                 // 50000 floats
    float* dinv = ws + 50048;         // 50000 floats (aligned stride)
    float* X    = ws + 100096;        // 50000*64 floats, 16-float aligned

    const int* rowIdx = edges;
    const int* colIdx = edges + N_EDGES;

    gcn_init_deg<<<(N_NODES + 255) / 256, 256, 0, stream>>>(deg);
    gcn_accum_deg<<<(N_EDGES + 255) / 256, 256, 0, stream>>>(colIdx, deg);
    gcn_dinv<<<(N_NODES + 255) / 256, 256, 0, stream>>>(deg, dinv);

    // 3125 row-blocks of 16 rows, 8 waves per 256-thread block
    gcn_gemm_wmma<<<(3125 + 7) / 8, 256, 0, stream>>>(features, W, X);

    gcn_self<<<(N_NODES * D + 255) / 256, 256, 0, stream>>>(X, dinv, bias, out);

    gcn_scatter<<<(N_EDGES * 16 + 255) / 256, 256, 0, stream>>>(rowIdx, colIdx, X, dinv, out);
}